// B3SplineUWT_34797825032656
// MI455X (gfx1250) — compile-verified
//
#include <hip/hip_runtime.h>
#include <hip/hip_bf16.h>

// ---------------------------------------------------------------------------
// Fused 3-level à-trous B3-spline UWT for MI455X (gfx1250, wave32).
//
// One workgroup = one 64x64 output tile of one batch image.
//   - stage 96x96 (tile + halo 16, reflect-indexed) region of x into LDS,
//     using GLOBAL_LOAD_ASYNC_TO_LDS_B32 (ASYNCcnt) when the toolchain
//     exposes the builtin, else plain load+ds_store
//   - level d=1,2,4: separable 5-tap conv done as banded-Toeplitz matmuls
//     with V_WMMA_F32_16X16X4_F32 (f32 in / f32 acc -> precision matches the
//     f32 reference), intermediates live in LDS (3 x ~50KB buffers)
//   - write w_j = c_{j-1} - c_j and final c_3 with b128 coalesced stores
// ---------------------------------------------------------------------------

typedef float v2f __attribute__((ext_vector_type(2)));
typedef float v8f __attribute__((ext_vector_type(8)));

typedef __attribute__((address_space(1))) int as1_int;
typedef __attribute__((address_space(3))) int as3_int;

#define IMG_H 1024
#define IMG_W 1024

constexpr int TILE  = 64;          // output tile
constexpr int HALO  = 16;          // rounded up from the true 14 for 16-alignment
constexpr int REGN  = TILE + 2 * HALO;   // 96: c0 region
constexpr int ROWS  = REGN + 16;   // 112: +guard rows so edge chunks stay in-bounds
constexpr int PITCH = ROWS + 1;    // 113 (odd) -> conflict-free column access
constexpr int BUFSZ = ROWS * PITCH;            // floats per LDS buffer
constexpr int SMEM_BYTES = 3 * BUFSZ * (int)sizeof(float);  // 151,872 B
constexpr int NCHUNK = (REGN / 16) * (REGN / 16);           // 36 16x16 chunks

#if defined(__has_builtin)
#if __has_builtin(__builtin_amdgcn_global_load_async_to_lds_b32)
#define UWT_HAVE_ASYNC_LDS 1
#endif
#if __has_builtin(__builtin_amdgcn_s_wait_asynccnt)
#define UWT_HAVE_WAIT_ASYNC 1
#endif
#endif

// Banded Toeplitz entry for the dilated 5-tap B3 kernel:
// band(delta) = w[delta/D] if delta in {0, D, 2D, 3D, 4D} else 0,
// with w = {1/16, 1/4, 3/8, 1/4, 1/16}.
template <int D>
__device__ __forceinline__ float band(int delta) {
  constexpr int LOG2D = (D == 1) ? 0 : (D == 2) ? 1 : 2;
  bool ok = (delta >= 0) && (delta <= 4 * D) && ((delta & (D - 1)) == 0);
  int  m  = (delta < 0 ? 0 : delta) >> LOG2D;            // 0..4
  float w = (m == 2) ? 0.375f : ((m & 1) ? 0.25f : 0.0625f);
  return ok ? w : 0.0f;
}

// One wavelet level: dst = conv_W(conv_H(src)) with dilation D.
// Buffer index convention: out[i][j] = sum_m sum_n w[m] w[n] src[i+m*D][j+n*D]
// (buffer origin shifts by 2*D per level; handled by caller offsets).
template <int D>
__device__ __forceinline__ void level_pass(const float* __restrict__ src,
                                           float* __restrict__ tmp,
                                           float* __restrict__ dst,
                                           int wave, int p, int h) {
  constexpr int K  = 16 + 4 * D;   // input span per 16-wide output chunk
  constexpr int NK = K / 4;        // WMMA k-steps (16x16x4)

  // Constant Toeplitz fragments (identical for both passes; A-layout for the
  // H pass, B-layout for the W pass -- both reduce to band(kglob - p)).
  v2f cf[NK];
#pragma unroll
  for (int s = 0; s < NK; ++s) {
    int kb = 4 * s + 2 * h;
    cf[s].x = band<D>(kb - p);
    cf[s].y = band<D>(kb + 1 - p);
  }

  // ---- H pass: tmp[i][j] = sum_m w[m] * src[i + m*D][j] ----
  // D(16x16) = T(16xK) x src_chunk(Kx16)
  for (int chunk = wave; chunk < NCHUNK; chunk += 8) {
    int r0 = (chunk / (REGN / 16)) * 16;
    int c0 = (chunk % (REGN / 16)) * 16;
    v8f acc = {};
#pragma unroll
    for (int s = 0; s < NK; ++s) {
      int kr = r0 + 4 * s + 2 * h;
      v2f bf;                                   // B-frag: rows kr, kr+1, col c0+p
      bf.x = src[kr * PITCH + c0 + p];
      bf.y = src[(kr + 1) * PITCH + c0 + p];
      acc = __builtin_amdgcn_wmma_f32_16x16x4_f32(
          false, cf[s], false, bf, (short)0, acc, false, false);
    }
#pragma unroll
    for (int v = 0; v < 8; ++v)                 // C layout: row = r0 + v + 8*h
      tmp[(r0 + 8 * h + v) * PITCH + c0 + p] = acc[v];
  }
  __syncthreads();

  // ---- W pass: dst[i][j] = sum_m w[m] * tmp[i][j + m*D] ----
  // D(16x16) = tmp_chunk(16xK) x T'(Kx16)
  for (int chunk = wave; chunk < NCHUNK; chunk += 8) {
    int r0 = (chunk / (REGN / 16)) * 16;
    int c0 = (chunk % (REGN / 16)) * 16;
    v8f acc = {};
#pragma unroll
    for (int s = 0; s < NK; ++s) {
      const float* q = &tmp[(r0 + p) * PITCH + c0 + 4 * s + 2 * h];
      v2f af;                                   // A-frag: row r0+p, 2 consecutive cols
      af.x = q[0];
      af.y = q[1];
      acc = __builtin_amdgcn_wmma_f32_16x16x4_f32(
          false, af, false, cf[s], (short)0, acc, false, false);
    }
#pragma unroll
    for (int v = 0; v < 8; ++v)
      dst[(r0 + 8 * h + v) * PITCH + c0 + p] = acc[v];
  }
  __syncthreads();
}

__global__ __launch_bounds__(256)
void B3SplineUWT_34797825032656_kernel(const float* __restrict__ x,
                                       float* __restrict__ out) {
  extern __shared__ float smem[];
  float* A  = smem;               // c_{j-1}
  float* T  = smem + BUFSZ;       // separable-conv temp
  float* Bn = smem + 2 * BUFSZ;   // c_j

  const int lane = threadIdx.x & 31;
  const int wave = threadIdx.x >> 5;
  const int p    = lane & 15;
  const int h    = lane >> 4;

  const int tx0 = (int)blockIdx.x * TILE - HALO;
  const int ty0 = (int)blockIdx.y * TILE - HALO;
  const float* xb = x + (size_t)blockIdx.z * IMG_H * IMG_W;

  // ---- stage c0 tile (+halo, reflect boundary) into LDS ----
  for (int idx = threadIdx.x; idx < REGN * REGN; idx += 256) {
    int r = idx / REGN, c = idx % REGN;
    int gy = ty0 + r;
    gy = gy < 0 ? -gy : (gy >= IMG_H ? 2 * IMG_H - 2 - gy : gy);
    int gx = tx0 + c;
    gx = gx < 0 ? -gx : (gx >= IMG_W ? 2 * IMG_W - 2 - gx : gx);
    const float* gp = xb + (size_t)gy * IMG_W + gx;
    float* lp = &A[r * PITCH + c];
#if defined(UWT_HAVE_ASYNC_LDS)
    // Direct global -> LDS DMA (no VGPR round-trip); tracked by ASYNCcnt.
    __builtin_amdgcn_global_load_async_to_lds_b32(
        (as1_int*)(uintptr_t)gp,
        (as3_int*)(uintptr_t)lp,
        0, 0);
#else
    *lp = *gp;
#endif
  }
#if defined(UWT_HAVE_ASYNC_LDS)
#if defined(UWT_HAVE_WAIT_ASYNC)
  __builtin_amdgcn_s_wait_asynccnt(0);
#else
  asm volatile("s_wait_asynccnt 0x0" ::: "memory");
#endif
#endif
  __syncthreads();

  // Buffer-origin offsets: abs y maps to buffer index y + off[j]
  // c0:+16, c1:+14, c2:+10, c3:+2  (origin shifts by 2*D each level)
  auto write_plane = [&](const float* prev, const float* next,
                         int po, int no, int plane) {
    float* ob = out + (((size_t)blockIdx.z * 4 + plane) * IMG_H +
                       (size_t)blockIdx.y * TILE) * IMG_W +
                (size_t)blockIdx.x * TILE;
    for (int g = threadIdx.x; g < TILE * TILE / 4; g += 256) {
      int yy = g >> 4;                 // 16 float4 groups per 64-wide row
      int xx = (g & 15) * 4;
      const float* pr = &prev[(yy + po) * PITCH + (xx + po)];
      const float* nx = &next[(yy + no) * PITCH + (xx + no)];
      float4 v;
      v.x = pr[0] - nx[0];
      v.y = pr[1] - nx[1];
      v.z = pr[2] - nx[2];
      v.w = pr[3] - nx[3];
      *(float4*)(ob + (size_t)yy * IMG_W + xx) = v;   // global_store_b128
    }
  };

  // ---- level 1 (d=1): w1 = c0 - c1 ----
  level_pass<1>(A, T, Bn, wave, p, h);
  write_plane(A, Bn, 16, 14, 0);
  __syncthreads();
  { float* t = A; A = Bn; Bn = t; }

  // ---- level 2 (d=2): w2 = c1 - c2 ----
  level_pass<2>(A, T, Bn, wave, p, h);
  write_plane(A, Bn, 14, 10, 1);
  __syncthreads();
  { float* t = A; A = Bn; Bn = t; }

  // ---- level 3 (d=4): w3 = c2 - c3, and the smooth plane c3 ----
  level_pass<4>(A, T, Bn, wave, p, h);
  write_plane(A, Bn, 10, 2, 2);
  {
    float* ob = out + (((size_t)blockIdx.z * 4 + 3) * IMG_H +
                       (size_t)blockIdx.y * TILE) * IMG_W +
                (size_t)blockIdx.x * TILE;
    for (int g = threadIdx.x; g < TILE * TILE / 4; g += 256) {
      int yy = g >> 4;
      int xx = (g & 15) * 4;
      const float* sm = &Bn[(yy + 2) * PITCH + (xx + 2)];
      float4 v;
      v.x = sm[0]; v.y = sm[1]; v.z = sm[2]; v.w = sm[3];
      *(float4*)(ob + (size_t)yy * IMG_W + xx) = v;
    }
  }
}

extern "C" void kernel_launch(void* const* d_in, const int* in_sizes, int n_in,
                              void* d_out, int out_size, void* d_ws, size_t ws_size,
                              hipStream_t stream) {
  const float* x = (const float*)d_in[0];
  float* out = (float*)d_out;
  int batch = in_sizes[0] / (IMG_H * IMG_W);   // 8 for the reference shapes
  dim3 grid(IMG_W / TILE, IMG_H / TILE, batch);
  B3SplineUWT_34797825032656_kernel<<<grid, dim3(256), SMEM_BYTES, stream>>>(x, out);
}